// CGCNN_44590350467111
// MI455X (gfx1250) — compile-verified
//
#include <hip/hip_runtime.h>
#include <hip/hip_bf16.h>
#include <math.h>

// ---------------- problem constants (match reference) ----------------
#define N_NODES  50000
#define N_EDGES  800000
#define N_GRAPHS 500
#define DIM      64
#define NG       64
#define CIN      192        // 2*DIM + NG
#define NCONV    3
#define HID      128
#define BN_EPS   1e-5f
#define CUTOFF   12.0f

typedef float v2f __attribute__((ext_vector_type(2)));
typedef float v8f __attribute__((ext_vector_type(8)));

__device__ __forceinline__ float sigmoidf_(float x) {
    return 1.0f / (1.0f + expf(-x));
}
__device__ __forceinline__ float softplusf_(float x) {
    // stable: max(x,0) + log1p(exp(-|x|))
    return fmaxf(x, 0.0f) + log1pf(expf(-fabsf(x)));
}

// ---------------------------------------------------------------------
// node init: embedding gather  node[n,c] = emb[atom_types[n], c]
// ---------------------------------------------------------------------
__global__ __launch_bounds__(256)
void k_init_node(const float* __restrict__ emb, const int* __restrict__ at,
                 float* __restrict__ node) {
    int idx = blockIdx.x * 256 + threadIdx.x;   // exactly N_NODES*DIM threads
    int n = idx >> 6, c = idx & 63;
    node[idx] = emb[at[n] * DIM + c];
}

// ---------------------------------------------------------------------
// fused edge conv pass (WMMA GEMM, recomputed per mode).
//  mode 0: accumulate per-column sum/sumsq of z = te@W + b
//  mode 1: f = sigmoid(bn1(z)); accumulate sum/sumsq of f
//  mode 2: h = f * softplus(bn2(f)); atomic scatter-add into agg[dst]
// block = 128 threads (4 waves), 64 edges; grid = E/64 (exact)
// ---------------------------------------------------------------------
#define LSTR 196   // padded LDS row stride (floats) -> conflict-free frags

__global__ __launch_bounds__(128)
void k_conv_pass(int mode,
                 const float* __restrict__ node,
                 const int*   __restrict__ src,
                 const int*   __restrict__ dst,
                 const float* __restrict__ dist,
                 const float* __restrict__ W,       // [192,64] row-major
                 const float* __restrict__ bias,    // [64]
                 const float* __restrict__ gamma1, const float* __restrict__ beta1,
                 const float* __restrict__ mean1,  const float* __restrict__ istd1,
                 const float* __restrict__ gamma2, const float* __restrict__ beta2,
                 const float* __restrict__ mean2,  const float* __restrict__ istd2,
                 float* __restrict__ Ssum, float* __restrict__ Sq,
                 float* __restrict__ agg) {
    __shared__ float te_s[64 * LSTR];      // 50176 B
    __shared__ int   es[64], ed[64];
    __shared__ float ds_[64];
    __shared__ float s_sum[64], s_sq[64];

    const int t     = threadIdx.x;
    const int ebase = blockIdx.x * 64;

    // stage edge indices + distances
    for (int j = t; j < 192; j += 128) {
        if (j < 64)       es[j]        = src[ebase + j];
        else if (j < 128) ed[j - 64]   = dst[ebase + j - 64];
        else              ds_[j - 128] = dist[ebase + j - 128];
    }
    if (t < 64) { s_sum[t] = 0.0f; s_sq[t] = 0.0f; }
    __syncthreads();

    // stage te tile: [node[src] | node[dst] | rbf(dist)]
    const float MUSTEP = CUTOFF / 63.0f;
    const float COEFF  = -0.5f / (MUSTEP * MUSTEP);
    for (int e = 0; e < 64; ++e) {
        const int   srow = es[e] * DIM;
        const int   drow = ed[e] * DIM;
        const float dd   = ds_[e];
        for (int c = t; c < CIN; c += 128) {
            float v;
            if (c < 64)       v = node[srow + c];
            else if (c < 128) v = node[drow + c - 64];
            else {
                float dm = dd - (float)(c - 128) * MUSTEP;
                v = expf(COEFF * dm * dm);
            }
            te_s[e * LSTR + c] = v;
        }
    }
    __syncthreads();

    // WMMA: wave w handles edges [16w,16w+16), all 4 N-tiles of 16 cols
    const int wave = t >> 5;
    const int lane = t & 31;
    const int half = lane >> 4;
    const int l16  = lane & 15;
    const int koff = 2 * half;
    const float* ta = &te_s[(wave * 16 + l16) * LSTR];

    for (int n = 0; n < 4; ++n) {
        const int col = n * 16 + l16;
        const float bsv = bias[col];
        v8f acc = {bsv, bsv, bsv, bsv, bsv, bsv, bsv, bsv};
        const float* wc = W + col;              // column `col` of W
        for (int k = 0; k < CIN; k += 4) {
            v2f a, b;
            a.x = ta[k + koff];
            a.y = ta[k + koff + 1];
            b.x = wc[(k + koff) * DIM];
            b.y = wc[(k + koff + 1) * DIM];
            acc = __builtin_amdgcn_wmma_f32_16x16x4_f32(
                false, a, false, b, (short)0, acc, false, false);
        }

        if (mode == 0) {
            float ls = 0.0f, lq = 0.0f;
#pragma unroll
            for (int r = 0; r < 8; ++r) { float z = acc[r]; ls += z; lq += z * z; }
            atomicAdd(&s_sum[col], ls);
            atomicAdd(&s_sq[col], lq);
        } else {
            const float m1 = mean1[col], i1 = istd1[col];
            const float g1 = gamma1[col], b1 = beta1[col];
            if (mode == 1) {
                float ls = 0.0f, lq = 0.0f;
#pragma unroll
                for (int r = 0; r < 8; ++r) {
                    float f = sigmoidf_(g1 * (acc[r] - m1) * i1 + b1);
                    ls += f; lq += f * f;
                }
                atomicAdd(&s_sum[col], ls);
                atomicAdd(&s_sq[col], lq);
            } else {
                const float m2 = mean2[col], i2 = istd2[col];
                const float g2 = gamma2[col], b2 = beta2[col];
#pragma unroll
                for (int r = 0; r < 8; ++r) {
                    float f   = sigmoidf_(g1 * (acc[r] - m1) * i1 + b1);
                    float arg = g2 * (f - m2) * i2 + b2;
                    float h   = f * softplusf_(arg);
                    int erow  = wave * 16 + r + 8 * half;   // D row = edge in tile
                    atomicAdd(&agg[ed[erow] * DIM + col], h);
                }
            }
        }
    }

    if (mode != 2) {
        __syncthreads();
        if (t < 64)        atomicAdd(&Ssum[t], s_sum[t]);
        else               atomicAdd(&Sq[t - 64], s_sq[t - 64]);
    }
}

// ---------------------------------------------------------------------
// per-column stats over agg [N_NODES, 64]
// ---------------------------------------------------------------------
__global__ __launch_bounds__(256)
void k_agg_stats(const float* __restrict__ agg,
                 float* __restrict__ Ssum, float* __restrict__ Sq) {
    __shared__ float ls[256], lq[256];
    int t = threadIdx.x;
    int c = t & 63, rg = t >> 6;
    int rowbase = blockIdx.x * 64;
    float s = 0.0f, q = 0.0f;
    for (int j = rg; j < 64; j += 4) {
        int r = rowbase + j;
        if (r < N_NODES) {
            float v = agg[r * DIM + c];
            s += v; q += v * v;
        }
    }
    ls[t] = s; lq[t] = q;
    __syncthreads();
    if (t < 64) {
        atomicAdd(&Ssum[t], ls[t] + ls[t + 64] + ls[t + 128] + ls[t + 192]);
    } else if (t < 128) {
        int c2 = t - 64;
        atomicAdd(&Sq[c2], lq[c2] + lq[c2 + 64] + lq[c2 + 128] + lq[c2 + 192]);
    }
}

// ---------------------------------------------------------------------
// finalize BN: mean/inv-std from sums, then reset accumulators
// ---------------------------------------------------------------------
__global__ void k_bn_finalize(float count,
                              float* __restrict__ Ssum, float* __restrict__ Sq,
                              float* __restrict__ mean, float* __restrict__ istd) {
    int t = threadIdx.x;
    if (t < 64) {
        float m = Ssum[t] / count;
        float v = Sq[t] / count - m * m;
        mean[t] = m;
        istd[t] = rsqrtf(v + BN_EPS);
        Ssum[t] = 0.0f;
        Sq[t]   = 0.0f;
    }
}

// ---------------------------------------------------------------------
// node += bn3(agg); also zero agg for next layer
// ---------------------------------------------------------------------
__global__ __launch_bounds__(256)
void k_node_update(float* __restrict__ node, float* __restrict__ agg,
                   const float* __restrict__ g3, const float* __restrict__ bt3,
                   const float* __restrict__ mean, const float* __restrict__ istd) {
    int idx = blockIdx.x * 256 + threadIdx.x;
    int c = idx & 63;
    float a = agg[idx];
    node[idx] += g3[c] * (a - mean[c]) * istd[c] + bt3[c];
    agg[idx] = 0.0f;
}

// ---------------------------------------------------------------------
// per-graph pooling (sum + counts)
// ---------------------------------------------------------------------
__global__ __launch_bounds__(256)
void k_pool(const float* __restrict__ node, const int* __restrict__ gid,
            float* __restrict__ psum, float* __restrict__ pcnt) {
    int idx = blockIdx.x * 256 + threadIdx.x;
    int n = idx >> 6, c = idx & 63;
    int g = gid[n];
    atomicAdd(&psum[g * DIM + c], node[idx]);
    if (c == 0) atomicAdd(&pcnt[g], 1.0f);
}

// ---------------------------------------------------------------------
// head MLP: softplus -> [64x128] fc -> softplus -> [128x1] out
// one block (128 thr) per graph
// ---------------------------------------------------------------------
__global__ __launch_bounds__(128)
void k_head(const float* __restrict__ psum, const float* __restrict__ pcnt,
            const float* __restrict__ Wfc, const float* __restrict__ bfc,
            const float* __restrict__ Wout, const float* __restrict__ bout,
            float* __restrict__ out) {
    __shared__ float sp[64];
    __shared__ float red[128];
    int g = blockIdx.x, t = threadIdx.x;
    if (t < 64) {
        float cnt = fmaxf(pcnt[g], 1.0f);
        sp[t] = softplusf_(psum[g * DIM + t] / cnt);
    }
    __syncthreads();
    float acc = bfc[t];
    for (int k = 0; k < 64; ++k) acc += sp[k] * Wfc[k * HID + t];
    red[t] = softplusf_(acc) * Wout[t];
    __syncthreads();
    for (int s = 64; s > 0; s >>= 1) {
        if (t < s) red[t] += red[t + s];
        __syncthreads();
    }
    if (t == 0) out[g] = red[0] + bout[0];
}

// ---------------------------------------------------------------------
// host orchestration
// ---------------------------------------------------------------------
extern "C" void kernel_launch(void* const* d_in, const int* in_sizes, int n_in,
                              void* d_out, int out_size, void* d_ws, size_t ws_size,
                              hipStream_t stream) {
    (void)in_sizes; (void)n_in; (void)out_size; (void)ws_size;

    const int*   atom_types = (const int*)  d_in[0];
    const int*   src        = (const int*)  d_in[1];
    const int*   dst        = (const int*)  d_in[2];
    const int*   graph_ids  = (const int*)  d_in[3];
    const float* distance   = (const float*)d_in[4];
    const float* embed      = (const float*)d_in[5];
    const float* W1         = (const float*)d_in[6];
    const float* b1         = (const float*)d_in[7];
    // d_in[8] = W2, d_in[9] = b2 : intentionally unused (faithful to source bug)
    const float* g1  = (const float*)d_in[10];
    const float* bt1 = (const float*)d_in[11];
    const float* g2  = (const float*)d_in[12];
    const float* bt2 = (const float*)d_in[13];
    const float* g3  = (const float*)d_in[14];
    const float* bt3 = (const float*)d_in[15];
    const float* Wfc = (const float*)d_in[16];
    const float* bfc = (const float*)d_in[17];
    const float* Wout = (const float*)d_in[18];
    const float* bout = (const float*)d_in[19];
    float* out = (float*)d_out;

    // workspace layout (all offsets 256B aligned)
    char* ws = (char*)d_ws;
    const size_t NODE_B = (size_t)N_NODES * DIM * sizeof(float);   // 12.8 MB
    float* node  = (float*)(ws);
    float* agg   = (float*)(ws + NODE_B);
    float* Ssum  = (float*)(ws + 2 * NODE_B);
    float* Sq    = (float*)(ws + 2 * NODE_B + 256);
    float* mean1 = (float*)(ws + 2 * NODE_B + 512);
    float* istd1 = (float*)(ws + 2 * NODE_B + 768);
    float* mean2 = (float*)(ws + 2 * NODE_B + 1024);
    float* istd2 = (float*)(ws + 2 * NODE_B + 1280);
    float* mean3 = (float*)(ws + 2 * NODE_B + 1536);
    float* istd3 = (float*)(ws + 2 * NODE_B + 1792);
    float* psum  = (float*)(ws + 2 * NODE_B + 2048);
    float* pcnt  = (float*)(ws + 2 * NODE_B + 2048 + (size_t)N_GRAPHS * DIM * sizeof(float));

    // deterministic zeroing every call (graph-capture safe)
    hipMemsetAsync(agg,  0, NODE_B, stream);
    hipMemsetAsync(Ssum, 0, 512, stream);                               // Ssum + Sq
    hipMemsetAsync(psum, 0, (size_t)N_GRAPHS * (DIM + 1) * sizeof(float), stream);

    const int NV_BLOCKS = (N_NODES * DIM) / 256;   // 12500
    const int E_BLOCKS  = N_EDGES / 64;            // 12500
    const int AS_BLOCKS = (N_NODES + 63) / 64;     // 782

    k_init_node<<<NV_BLOCKS, 256, 0, stream>>>(embed, atom_types, node);

    for (int i = 0; i < NCONV; ++i) {
        const float* Wl   = W1 + (size_t)i * CIN * DIM;
        const float* bl   = b1 + i * DIM;
        const float* g1l  = g1 + i * DIM;  const float* bt1l = bt1 + i * DIM;
        const float* g2l  = g2 + i * DIM;  const float* bt2l = bt2 + i * DIM;
        const float* g3l  = g3 + i * DIM;  const float* bt3l = bt3 + i * DIM;

        // pass 1: stats of z
        k_conv_pass<<<E_BLOCKS, 128, 0, stream>>>(0, node, src, dst, distance,
            Wl, bl, g1l, bt1l, mean1, istd1, g2l, bt2l, mean2, istd2,
            Ssum, Sq, agg);
        k_bn_finalize<<<1, 64, 0, stream>>>((float)N_EDGES, Ssum, Sq, mean1, istd1);

        // pass 2: stats of f = sigmoid(bn1(z))
        k_conv_pass<<<E_BLOCKS, 128, 0, stream>>>(1, node, src, dst, distance,
            Wl, bl, g1l, bt1l, mean1, istd1, g2l, bt2l, mean2, istd2,
            Ssum, Sq, agg);
        k_bn_finalize<<<1, 64, 0, stream>>>((float)N_EDGES, Ssum, Sq, mean2, istd2);

        // pass 3: h = f * softplus(bn2(f)); scatter to agg
        k_conv_pass<<<E_BLOCKS, 128, 0, stream>>>(2, node, src, dst, distance,
            Wl, bl, g1l, bt1l, mean1, istd1, g2l, bt2l, mean2, istd2,
            Ssum, Sq, agg);

        // bn3 over agg, then node update (also re-zeroes agg)
        k_agg_stats<<<AS_BLOCKS, 256, 0, stream>>>(agg, Ssum, Sq);
        k_bn_finalize<<<1, 64, 0, stream>>>((float)N_NODES, Ssum, Sq, mean3, istd3);
        k_node_update<<<NV_BLOCKS, 256, 0, stream>>>(node, agg, g3l, bt3l, mean3, istd3);
    }

    k_pool<<<NV_BLOCKS, 256, 0, stream>>>(node, graph_ids, psum, pcnt);
    k_head<<<N_GRAPHS, 128, 0, stream>>>(psum, pcnt, Wfc, bfc, Wout, bout, out);
}